// SSPNet_2310692405650
// MI455X (gfx1250) — compile-verified
//
#include <hip/hip_runtime.h>
#include <math.h>

typedef __attribute__((ext_vector_type(2))) float v2f;
typedef __attribute__((ext_vector_type(8))) float v8f;

#define F_IN 256
#define HID  128
#define NCLS 40

// ---------------- degree / normalization ----------------

__global__ void k_deg_init(float* __restrict__ deg, int n) {
    int i = blockIdx.x * blockDim.x + threadIdx.x;
    if (i < n) deg[i] = 1.0f;   // self-loop contributes 1 to every node's degree
}

__global__ void k_deg_accum(const int* __restrict__ dst, float* __restrict__ deg, int e) {
    int i = blockIdx.x * blockDim.x + threadIdx.x;
    if (i < e) atomicAdd(&deg[dst[i]], 1.0f);
}

__global__ void k_rsqrt_inplace(float* __restrict__ d, int n) {
    int i = blockIdx.x * blockDim.x + threadIdx.x;
    if (i < n) d[i] = rsqrtf(d[i]);   // deg >= 1 always (self-loops)
}

__global__ void k_edge_norm(const int* __restrict__ src, const int* __restrict__ dst,
                            const float* __restrict__ dinv, float* __restrict__ norm, int e) {
    int i = blockIdx.x * blockDim.x + threadIdx.x;
    if (i < e) norm[i] = dinv[src[i]] * dinv[dst[i]];
}

// ---------------- fp32 WMMA GEMMs ----------------
// Layer 1: xw[N,128] = x[N,256] @ W1[256,128]; 8 waves/block, one 16x16 tile each.
__global__ __launch_bounds__(256) void k_gemm1(const float* __restrict__ x,
                                               const float* __restrict__ W1,
                                               float* __restrict__ xw, int n) {
    __shared__ float lx[16 * F_IN];
    int rowBase = blockIdx.x * 16;
    if (rowBase >= n) return;
    int rows = min(16, n - rowBase);
    for (int idx = threadIdx.x; idx < 16 * F_IN; idx += 256)
        lx[idx] = (idx < rows * F_IN) ? x[(size_t)rowBase * F_IN + idx] : 0.0f;
    __syncthreads();

    int wave = threadIdx.x >> 5;
    int lane = threadIdx.x & 31;
    int half = lane >> 4;      // 0: K pair {0,1};  1: K pair {2,3}
    int l    = lane & 15;      // M for A-frag, N for B/C-frag
    int nBase = wave * 16;
    const float* lrow = &lx[l * F_IN];

    v8f c = {};
    for (int k = 0; k < F_IN; k += 4) {
        v2f a, b;
        a.x = lrow[k + 2 * half];
        a.y = lrow[k + 2 * half + 1];
        int ncol = nBase + l;
        b.x = W1[(k + 2 * half) * HID + ncol];
        b.y = W1[(k + 2 * half + 1) * HID + ncol];
        c = __builtin_amdgcn_wmma_f32_16x16x4_f32(false, a, false, b,
                                                  (short)0, c, false, false);
    }
    int ncol = nBase + l;
    float* outp = xw + (size_t)(rowBase + half * 8) * HID + ncol;
    if (rows == 16) {
#pragma unroll
        for (int r = 0; r < 8; ++r) outp[(size_t)r * HID] = c[r];
    } else {
        int m0 = half * 8;
        for (int r = 0; r < 8; ++r)
            if (m0 + r < rows) outp[(size_t)r * HID] = c[r];
    }
}

// Layer 2: hw[N,40] = z[N,128] @ W2[128,40]; 3 waves/block (tiles cover cols 0..47, pad >=40 with 0).
__global__ __launch_bounds__(96) void k_gemm2(const float* __restrict__ z,
                                              const float* __restrict__ W2,
                                              float* __restrict__ hw, int n) {
    __shared__ float lz[16 * HID];
    int rowBase = blockIdx.x * 16;
    if (rowBase >= n) return;
    int rows = min(16, n - rowBase);
    for (int idx = threadIdx.x; idx < 16 * HID; idx += 96)
        lz[idx] = (idx < rows * HID) ? z[(size_t)rowBase * HID + idx] : 0.0f;
    __syncthreads();

    int wave = threadIdx.x >> 5;
    int lane = threadIdx.x & 31;
    int half = lane >> 4;
    int l    = lane & 15;
    int ncol = wave * 16 + l;
    const float* lrow = &lz[l * HID];

    v8f c = {};
    for (int k = 0; k < HID; k += 4) {
        v2f a, b;
        a.x = lrow[k + 2 * half];
        a.y = lrow[k + 2 * half + 1];
        if (ncol < NCLS) {
            b.x = W2[(k + 2 * half) * NCLS + ncol];
            b.y = W2[(k + 2 * half + 1) * NCLS + ncol];
        } else {
            b.x = 0.0f; b.y = 0.0f;
        }
        c = __builtin_amdgcn_wmma_f32_16x16x4_f32(false, a, false, b,
                                                  (short)0, c, false, false);
    }
    if (ncol < NCLS) {
        float* outp = hw + (size_t)(rowBase + half * 8) * NCLS + ncol;
        if (rows == 16) {
#pragma unroll
            for (int r = 0; r < 8; ++r) outp[(size_t)r * NCLS] = c[r];
        } else {
            int m0 = half * 8;
            for (int r = 0; r < 8; ++r)
                if (m0 + r < rows) outp[(size_t)r * NCLS] = c[r];
        }
    }
}

// ---------------- aggregation (self-loop init + edge scatter-add) ----------------
// All feature dims are multiples of 4 -> float4 vectorized (16B-aligned offsets).

template <int F>
__global__ void k_selfloop_init(const float* __restrict__ dinv, const float* __restrict__ v,
                                float* __restrict__ agg, int n) {
    constexpr int C = F / 4;
    int idx = blockIdx.x * blockDim.x + threadIdx.x;   // over n*C float4 chunks
    if (idx >= n * C) return;
    int i = idx / C;
    float d = dinv[i];
    float w = d * d;
    float4 val = ((const float4*)v)[idx];
    val.x *= w; val.y *= w; val.z *= w; val.w *= w;
    ((float4*)agg)[idx] = val;
}

template <int F>
__global__ void k_edge_scatter(const int* __restrict__ src, const int* __restrict__ dst,
                               const float* __restrict__ norm, const float* __restrict__ v,
                               float* __restrict__ agg, int e) {
    constexpr int C = F / 4;                            // float4 chunks per edge
    long long idx = (long long)blockIdx.x * blockDim.x + threadIdx.x;
    long long total = (long long)e * C;
    if (idx >= total) return;
    int eid = (int)(idx / C);
    int ch  = (int)(idx - (long long)eid * C);
    int s = src[eid], d = dst[eid];
    float w = norm[eid];
    float4 val = *(const float4*)(v + (size_t)s * F + ch * 4);
    float* ap = agg + (size_t)d * F + ch * 4;
    atomicAdd(ap + 0, w * val.x);
    atomicAdd(ap + 1, w * val.y);
    atomicAdd(ap + 2, w * val.z);
    atomicAdd(ap + 3, w * val.w);
}

__global__ void k_bias_relu(const float* __restrict__ agg, const float* __restrict__ b,
                            float* __restrict__ out, int n) {
    constexpr int C = HID / 4;
    int idx = blockIdx.x * blockDim.x + threadIdx.x;   // over n*C float4 chunks
    if (idx >= n * C) return;
    int ch = idx & (C - 1);
    float4 val = ((const float4*)agg)[idx];
    const float4 bb = ((const float4*)b)[ch];
    val.x = fmaxf(val.x + bb.x, 0.0f);
    val.y = fmaxf(val.y + bb.y, 0.0f);
    val.z = fmaxf(val.z + bb.z, 0.0f);
    val.w = fmaxf(val.w + bb.w, 0.0f);
    ((float4*)out)[idx] = val;
}

// One wave32 per row: lanes cover logits [lane] and [lane+32] (NCLS=40).
__global__ __launch_bounds__(256) void k_logsoftmax(const float* __restrict__ agg,
                                                    const float* __restrict__ b,
                                                    float* __restrict__ out, int n) {
    int wave = threadIdx.x >> 5;
    int lane = threadIdx.x & 31;
    int row = blockIdx.x * 8 + wave;
    if (row >= n) return;
    const float* a = agg + (size_t)row * NCLS;
    float v0 = a[lane] + b[lane];
    float v1 = (lane < NCLS - 32) ? (a[32 + lane] + b[32 + lane]) : -INFINITY;
    float m = fmaxf(v0, v1);
#pragma unroll
    for (int off = 16; off; off >>= 1) m = fmaxf(m, __shfl_xor(m, off, 32));
    float s = expf(v0 - m) + ((lane < NCLS - 32) ? expf(v1 - m) : 0.0f);
#pragma unroll
    for (int off = 16; off; off >>= 1) s += __shfl_xor(s, off, 32);
    float ls = logf(s);
    out[(size_t)row * NCLS + lane] = v0 - m - ls;
    if (lane < NCLS - 32) out[(size_t)row * NCLS + 32 + lane] = v1 - m - ls;
}

// ---------------- driver ----------------

extern "C" void kernel_launch(void* const* d_in, const int* in_sizes, int n_in,
                              void* d_out, int out_size, void* d_ws, size_t ws_size,
                              hipStream_t stream) {
    const float* x  = (const float*)d_in[0];
    const int*   ei = (const int*)d_in[1];
    const float* W1 = (const float*)d_in[2];
    const float* b1 = (const float*)d_in[3];
    const float* W2 = (const float*)d_in[4];
    const float* b2 = (const float*)d_in[5];

    int N = in_sizes[0] / F_IN;
    int E = in_sizes[1] / 2;
    const int* src = ei;
    const int* dst = ei + E;

    float* ws   = (float*)d_ws;
    float* dinv = ws;                          // N   (degree, then rsqrt in place)
    float* norm = ws + N;                      // E
    float* bufA = ws + (size_t)N + E;          // 128*N (xw, then hw[N,40])
    float* bufB = bufA + (size_t)N * HID;      // 128*N (agg1, then agg2)

    float* zout  = (float*)d_out;                       // N*HID
    float* lsout = (float*)d_out + (size_t)N * HID;     // N*NCLS

    const int T = 256;
    // normalization
    k_deg_init<<<(N + T - 1) / T, T, 0, stream>>>(dinv, N);
    k_deg_accum<<<(E + T - 1) / T, T, 0, stream>>>(dst, dinv, E);
    k_rsqrt_inplace<<<(N + T - 1) / T, T, 0, stream>>>(dinv, N);
    k_edge_norm<<<(E + T - 1) / T, T, 0, stream>>>(src, dst, dinv, norm, E);

    // layer 1: GCNConv + relu
    k_gemm1<<<(N + 15) / 16, 256, 0, stream>>>(x, W1, bufA, N);
    k_selfloop_init<HID><<<(N * (HID / 4) + T - 1) / T, T, 0, stream>>>(dinv, bufA, bufB, N);
    {
        long long total = (long long)E * (HID / 4);
        int blocks = (int)((total + T - 1) / T);
        k_edge_scatter<HID><<<blocks, T, 0, stream>>>(src, dst, norm, bufA, bufB, E);
    }
    k_bias_relu<<<(N * (HID / 4) + T - 1) / T, T, 0, stream>>>(bufB, b1, zout, N);

    // layer 2: GCNConv + log_softmax
    k_gemm2<<<(N + 15) / 16, 96, 0, stream>>>(zout, W2, bufA, N);
    k_selfloop_init<NCLS><<<(N * (NCLS / 4) + T - 1) / T, T, 0, stream>>>(dinv, bufA, bufB, N);
    {
        long long total = (long long)E * (NCLS / 4);
        int blocks = (int)((total + T - 1) / T);
        k_edge_scatter<NCLS><<<blocks, T, 0, stream>>>(src, dst, norm, bufA, bufB, E);
    }
    k_logsoftmax<<<(N + 7) / 8, 256, 0, stream>>>(bufB, b2, lsout, N);
}